// WordRep_2628519985196
// MI455X (gfx1250) — compile-verified
//
#include <hip/hip_runtime.h>
#include <math.h>

// Problem constants (match reference)
#define BB 8
#define SS 512
#define DD 512
#define VV 100000
#define LL 128
#define ROWS (BB * SS)        // 4096
#define EPS 1e-8f

typedef __attribute__((ext_vector_type(2))) float v2f;
typedef __attribute__((ext_vector_type(8))) float v8f;

// ---------------------------------------------------------------------------
// Kernel 1: gather word embeddings, write both output copies, compute row norm
// One block (256 threads) per row; each thread handles 2 consecutive floats.
// ---------------------------------------------------------------------------
__global__ __launch_bounds__(256) void gather_and_norm(
    const int* __restrict__ word_inputs,
    const float* __restrict__ word_emb_table,
    float* __restrict__ out_rep,     // [4096, 512]
    float* __restrict__ out_orig,    // [4096, 512]
    float* __restrict__ wnorm)       // [4096]
{
    const int r = blockIdx.x;        // 0..4095
    const int t = threadIdx.x;       // 0..255
    const int idx = word_inputs[r];

    const float2* src = (const float2*)(word_emb_table + (size_t)idx * DD);
    float2 v = src[t];

    ((float2*)(out_rep  + (size_t)r * DD))[t] = v;
    ((float2*)(out_orig + (size_t)r * DD))[t] = v;

    __shared__ float red[256];
    red[t] = v.x * v.x + v.y * v.y;
    __syncthreads();
#pragma unroll
    for (int off = 128; off > 0; off >>= 1) {
        if (t < off) red[t] += red[t + off];
        __syncthreads();
    }
    if (t == 0) wnorm[r] = sqrtf(red[0]);
}

// ---------------------------------------------------------------------------
// Kernel 2: label row norms. One block per label row.
// ---------------------------------------------------------------------------
__global__ __launch_bounds__(256) void label_norm(
    const float* __restrict__ label_emb_table,  // [128, 512]
    float* __restrict__ lnorm)                  // [128]
{
    const int r = blockIdx.x;        // 0..127
    const int t = threadIdx.x;
    const float2* src = (const float2*)(label_emb_table + (size_t)r * DD);
    float2 v = src[t];

    __shared__ float red[256];
    red[t] = v.x * v.x + v.y * v.y;
    __syncthreads();
#pragma unroll
    for (int off = 128; off > 0; off >>= 1) {
        if (t < off) red[t] += red[t + off];
        __syncthreads();
    }
    if (t == 0) lnorm[r] = sqrtf(red[0]);
}

// ---------------------------------------------------------------------------
// Kernel 3: LS_embs[m,n] = (A[m,:] . L[n,:]) / max(wnorm[m]*lnorm[n], EPS)
// via V_WMMA_F32_16X16X4_F32.  One workgroup (8 waves) per 16-row M tile.
// A tile (16 x 512) staged in LDS with padded stride (516 floats) so the
// per-step ds_load_b64 A-fragment fetch is bank-conflict free (bank = 4*m).
// Each wave owns one 16-column N tile; B streams from L2 (label table is
// only 256 KB, far below the 192 MB L2).
// ---------------------------------------------------------------------------
#define LDSTRIDE 516   // 512 + 4 floats padding

__global__ __launch_bounds__(256) void ls_wmma_gemm(
    const float* __restrict__ A,      // gathered word_embs [4096, 512]
    const float* __restrict__ Lt,     // label_emb_table    [128, 512]
    const float* __restrict__ wnorm,  // [4096]
    const float* __restrict__ lnorm,  // [128]
    float* __restrict__ out)          // LS_embs [4096, 128]
{
    __shared__ float lds_a[16 * LDSTRIDE];   // 33024 bytes

    const int mtile = blockIdx.x;            // 0..255
    const int tid   = threadIdx.x;           // 0..255

    // ---- Stage A tile (16 rows x 512) into LDS, padded rows. ----
    const float4* asrc = (const float4*)(A + (size_t)mtile * 16 * DD);
    for (int i = tid; i < 16 * (DD / 4); i += 256) {
        const int row = i >> 7;              // /128 float4 per row
        const int col = i & 127;
        *((float4*)(lds_a + row * LDSTRIDE) + col) = asrc[(size_t)row * (DD / 4) + col];
    }
    __syncthreads();

    // ---- Per-lane WMMA operand addressing (ISA 7.12.2 layouts). ----
    const int wave  = tid >> 5;              // 0..7 -> N tile
    const int lane  = tid & 31;
    const int lhalf = lane >> 4;             // 0: K pair {0,1}; 1: K pair {2,3}
    const int l16   = lane & 15;             // M (for A) / N (for B)
    const int n     = wave * 16 + l16;       // global column (label id)

    const float* arow = lds_a + l16 * LDSTRIDE + 2 * lhalf;   // A[m=l16][k0+2h ..]
    const float* brow = Lt + (size_t)n * DD + 2 * lhalf;      // B[k][n] = L[n][k]

    v8f c = {};
#pragma unroll 8
    for (int k0 = 0; k0 < DD; k0 += 4) {
        v2f a = *(const v2f*)(arow + k0);    // K = k0+2h, k0+2h+1 for row l16
        v2f b = *(const v2f*)(brow + k0);
        // 8 args: (neg_a, A, neg_b, B, c_mod, C, reuse_a, reuse_b)
        c = __builtin_amdgcn_wmma_f32_16x16x4_f32(
                false, a, false, b, (short)0, c, false, false);
    }

    // ---- Fused cosine normalization + writeback.
    // C/D layout: VGPR v, lanes 0-15 -> M = v, lanes 16-31 -> M = v + 8.
    const float ln = lnorm[n];
#pragma unroll
    for (int v = 0; v < 8; ++v) {
        const int m = mtile * 16 + lhalf * 8 + v;
        const float denom = fmaxf(wnorm[m] * ln, EPS);
        out[(size_t)m * LL + n] = c[v] / denom;
    }
}

// ---------------------------------------------------------------------------
// Launch: out = [word_represent | LS_embs | orig_word_embs] flat.
// ---------------------------------------------------------------------------
extern "C" void kernel_launch(void* const* d_in, const int* in_sizes, int n_in,
                              void* d_out, int out_size, void* d_ws, size_t ws_size,
                              hipStream_t stream) {
    const int*   word_inputs     = (const int*)d_in[0];
    const float* word_emb_table  = (const float*)d_in[5];
    const float* label_emb_table = (const float*)d_in[6];

    float* out_rep  = (float*)d_out;                       // [4096*512]
    float* out_ls   = out_rep + (size_t)ROWS * DD;         // [4096*128]
    float* out_orig = out_ls + (size_t)ROWS * LL;          // [4096*512]

    float* wnorm = (float*)d_ws;                           // [4096]
    float* lnorm = wnorm + ROWS;                           // [128]

    gather_and_norm<<<ROWS, 256, 0, stream>>>(
        word_inputs, word_emb_table, out_rep, out_orig, wnorm);

    label_norm<<<LL, 256, 0, stream>>>(label_emb_table, lnorm);

    ls_wmma_gemm<<<ROWS / 16, 256, 0, stream>>>(
        out_rep, label_emb_table, wnorm, lnorm, out_ls);
}